// GCN_48490180772308
// MI455X (gfx1250) — compile-verified
//
#include <hip/hip_runtime.h>
#include <hip/hip_bf16.h>

// Problem constants (fixed by the reference setup)
#define BDIM   16
#define HDIM   8
#define RDIM   400
#define FDIM   256
#define MTOT   (BDIM * HDIM * RDIM)   // 51200 rows, divisible by 16
#define NSLOT  8                      // up to 6 rank-selected cols + diag + pad

typedef __attribute__((ext_vector_type(2))) float v2f;
typedef __attribute__((ext_vector_type(8))) float v8f;

// ---------------------------------------------------------------------------
// Kernel 1: per (b,h,i) row of A, compute the ranks (descending, stable) of
// columns {0, 8..12}; the mask selects exactly those rank positions plus the
// diagonal. Emit 8 (col, weight) slots per row: weight = A[row, rank_c].
// One wave32 per row.
// ---------------------------------------------------------------------------
__global__ void __launch_bounds__(256)
gcn_rank_select_kernel(const float* __restrict__ A,
                       int* __restrict__ wcols,
                       float* __restrict__ wvals) {
    const int m    = (blockIdx.x * blockDim.x + threadIdx.x) >> 5; // row id
    const int lane = threadIdx.x & 31;
    if (m >= MTOT) return;

    const int   i   = m % RDIM;
    const float* row = A + (size_t)m * RDIM;

    const int pivcol[6] = {0, 8, 9, 10, 11, 12};
    float pv[6];
#pragma unroll
    for (int c = 0; c < 6; ++c) pv[c] = row[pivcol[c]];

    // rank(c) = #{ j : A[j] > A[c]  ||  (A[j] == A[c] && j < c) }   (stable sort)
    int cnt[6] = {0, 0, 0, 0, 0, 0};
    for (int j = lane; j < RDIM; j += 32) {
        const float v = row[j];
#pragma unroll
        for (int c = 0; c < 6; ++c) {
            const bool before = (v > pv[c]) | ((v == pv[c]) & (j < pivcol[c]));
            cnt[c] += before ? 1 : 0;
        }
    }
#pragma unroll
    for (int c = 0; c < 6; ++c) {
#pragma unroll
        for (int off = 16; off > 0; off >>= 1)
            cnt[c] += __shfl_xor(cnt[c], off, 32);
    }

    if (lane == 0) {
        const size_t base = (size_t)m * NSLOT;
        bool dup = false;
#pragma unroll
        for (int s = 0; s < 6; ++s) dup = dup || (cnt[s] == i);
#pragma unroll
        for (int s = 0; s < 6; ++s) {
            const int k = cnt[s];            // selected column = rank position
            wcols[base + s] = k;
            wvals[base + s] = row[k];        // weight is A[i, k]
        }
        wcols[base + 6] = dup ? 0 : i;       // diagonal (dedup'd)
        wvals[base + 6] = dup ? 0.0f : row[i];
        wcols[base + 7] = 0;                 // pad slot contributes 0
        wvals[base + 7] = 0.0f;
    }
}

// ---------------------------------------------------------------------------
// Kernel 2: fused sparse-gather agg + fp32 WMMA GEMM + relu/bias/residual.
// One block = 512 threads (16 wave32) per 16-row M tile.
//   Phase 1: wave w gathers agg row w (<=8 weighted feature rows) into LDS.
//   Phase 2: wave w owns N-tile [16w, 16w+16); 64 x v_wmma_f32_16x16x4_f32
//            over K=256. A-frag from LDS, B-frag (=W^T) from global (L2-hot).
// LDS row stride 258 floats so the K-strided ds_load_b64 reads hit distinct banks.
// ---------------------------------------------------------------------------
#define LDS_STRIDE 258

__global__ void __launch_bounds__(512)
gcn_fused_gemm_kernel(const float* __restrict__ feats,
                      const float* __restrict__ W,     // [N=256, K=256] torch layout
                      const float* __restrict__ bias,  // [256]
                      const int*   __restrict__ wcols,
                      const float* __restrict__ wvals,
                      float* __restrict__ out) {
    __shared__ float sAgg[16 * LDS_STRIDE];

    const int tile = blockIdx.x;        // 0..3199
    const int tid  = threadIdx.x;
    const int wid  = tid >> 5;          // 0..15
    const int lane = tid & 31;

    // ---- Phase 1: build agg tile [16 x 256] in LDS ----
    {
        const int m       = tile * 16 + wid;
        const int rowbase = (m / RDIM) * RDIM;   // (b*H+h)*R
        const int*   cp = wcols + (size_t)m * NSLOT;
        const float* vp = wvals + (size_t)m * NSLOT;
        int   cofs[NSLOT];
        float cw[NSLOT];
#pragma unroll
        for (int s = 0; s < NSLOT; ++s) {
            cofs[s] = (rowbase + cp[s]) * FDIM;
            cw[s]   = vp[s];
        }
#pragma unroll
        for (int f = lane; f < FDIM; f += 32) {
            float acc = 0.0f;
#pragma unroll
            for (int s = 0; s < NSLOT; ++s)
                acc += cw[s] * feats[(size_t)cofs[s] + f];
            sAgg[wid * LDS_STRIDE + f] = acc;
        }
    }
    __syncthreads();

    // ---- Phase 2: WMMA over K=256 ----
    // A 16x4 f32 layout: lanes 0-15 hold M=lane, K = k0+{0,1};
    //                    lanes 16-31 hold M=lane-16, K = k0+{2,3}.
    // B 4x16 f32 layout mirrors A: lanes 0-15 -> K rows {0,1}, lanes 16-31 -> {2,3},
    //                    column N = n0 + (lane & 15).
    const int n0    = wid * 16;
    const int nn    = n0 + (lane & 15);
    const int khalf = (lane >> 4) * 2;
    const int mloc  = lane & 15;

    v8f acc = {};
    const float* wrow = W + (size_t)nn * FDIM;   // W^T(k, nn) = W[nn, k]
    for (int k0 = 0; k0 < FDIM; k0 += 4) {
        const v2f a = *(const v2f*)(&sAgg[mloc * LDS_STRIDE + k0 + khalf]);
        const v2f b = *(const v2f*)(&wrow[k0 + khalf]);
        acc = __builtin_amdgcn_wmma_f32_16x16x4_f32(
            /*neg_a=*/false, a, /*neg_b=*/false, b,
            /*c_mod=*/(short)0, acc, /*reuse_a=*/false, /*reuse_b=*/false);
    }

    // ---- Epilogue: relu(acc + bias) + features ----
    // C/D 16x16 f32 layout: VGPR v -> M = v + 8*(lane>=16); N = lane & 15.
    const float bn    = bias[nn];
    const int   mhalf = (lane >> 4) * 8;
#pragma unroll
    for (int v = 0; v < 8; ++v) {
        const int    m   = tile * 16 + mhalf + v;
        const size_t idx = (size_t)m * FDIM + nn;
        float val = acc[v] + bn;
        val = val > 0.0f ? val : 0.0f;
        out[idx] = val + feats[idx];
    }
}

// ---------------------------------------------------------------------------
extern "C" void kernel_launch(void* const* d_in, const int* in_sizes, int n_in,
                              void* d_out, int out_size, void* d_ws, size_t ws_size,
                              hipStream_t stream) {
    const float* A     = (const float*)d_in[0];   // [16,8,400,400]
    const float* feats = (const float*)d_in[1];   // [16,8,400,256]
    const float* W     = (const float*)d_in[2];   // [256,256]
    const float* bias  = (const float*)d_in[3];   // [256]
    float*       out   = (float*)d_out;           // [16,8,400,256]

    // Workspace: cols [MTOT][8] int32, then vals [MTOT][8] f32  (~3.3 MB)
    int*   wcols = (int*)d_ws;
    float* wvals = (float*)(wcols + (size_t)MTOT * NSLOT);

    // Kernel 1: one wave per row, 8 waves per block
    {
        const int waves_per_block = 8;
        const int blocks = MTOT / waves_per_block;   // 6400
        gcn_rank_select_kernel<<<blocks, waves_per_block * 32, 0, stream>>>(A, wcols, wvals);
    }

    // Kernel 2: one block per 16-row M tile
    {
        const int blocks = MTOT / 16;                // 3200
        gcn_fused_gemm_kernel<<<blocks, 512, 0, stream>>>(feats, W, bias, wcols, wvals, out);
    }
}